// GCN_12644383719572
// MI455X (gfx1250) — compile-verified
//
#include <hip/hip_runtime.h>

#define NN 100000
#define NE 1200000
#define DD 64
#define NREL 50
#define WSTRIDE 80   // LDS row stride in dwords: 80 % 64 == 16 -> conflict-free B reads

typedef __attribute__((ext_vector_type(2))) float v2f;
typedef __attribute__((ext_vector_type(8))) float v8f;
typedef __attribute__((ext_vector_type(4))) unsigned int v4u;
typedef __attribute__((ext_vector_type(8))) int v8i;
typedef __attribute__((ext_vector_type(4))) int v4i;

__device__ __forceinline__ void atomic_add_f32(float* p, float v) {
    __hip_atomic_fetch_add(p, v, __ATOMIC_RELAXED, __HIP_MEMORY_SCOPE_AGENT);
}

__global__ void zero_kernel(float* __restrict__ p, int n) {
    int i = blockIdx.x * blockDim.x + threadIdx.x;
    if (i < n) p[i] = 0.0f;
}

__global__ void degree_kernel(const int* __restrict__ src, const int* __restrict__ dst,
                              float* __restrict__ deg_out, float* __restrict__ deg_in) {
    int e = blockIdx.x * blockDim.x + threadIdx.x;
    if (e < NE) {
        atomic_add_f32(&deg_out[src[e]], 1.0f);
        atomic_add_f32(&deg_in[dst[e]], 1.0f);
    }
}

__global__ void norm_kernel(const float* __restrict__ deg_out, const float* __restrict__ deg_in,
                            float* __restrict__ norm_out, float* __restrict__ inv_in,
                            float* __restrict__ rsq_in) {
    int i = blockIdx.x * blockDim.x + threadIdx.x;
    if (i < NN) {
        norm_out[i] = rsqrtf(fmaxf(deg_out[i], 1.0f));
        float fi = fmaxf(deg_in[i], 1.0f);
        inv_in[i] = 1.0f / fi;
        rsq_in[i] = rsqrtf(fi);
    }
}

// One wave per edge: 32 lanes x 2 floats = 64-float message.
// m = x[src] * norm_out[src] * rel_weight[etype]; atomic scatter-add into msum[dst].
__global__ void __launch_bounds__(256)
edge_kernel(const float* __restrict__ x, const float* __restrict__ rel,
            const int* __restrict__ src, const int* __restrict__ dst,
            const int* __restrict__ etype, const float* __restrict__ norm_out,
            float* __restrict__ msum) {
    int wave = blockIdx.x * (blockDim.x >> 5) + (threadIdx.x >> 5);
    int lane = threadIdx.x & 31;
    if (wave >= NE) return;
    int s = src[wave];
    int d = dst[wave];
    int t = etype[wave];
    float sc = norm_out[s];
    v2f xv = *(const v2f*)(x + (size_t)s * DD + lane * 2);
    v2f rw = *(const v2f*)(rel + (size_t)t * DD + lane * 2);
    float* o = msum + (size_t)d * DD + lane * 2;
    atomic_add_f32(o + 0, xv.x * sc * rw.x);
    atomic_add_f32(o + 1, xv.y * sc * rw.y);
}

// One wave per 16-row node tile: out[16x64] = relu((x[16x64] @ W[64x64] + msum*inv_in) * rsq_in)
// W is staged into LDS once per workgroup by the Tensor Data Mover (with row padding for
// bank-conflict-free B-fragment reads), then consumed by V_WMMA_F32_16X16X4_F32.
__global__ void __launch_bounds__(256)
node_kernel(const float* __restrict__ x, const float* __restrict__ W,
            const float* __restrict__ msum, const float* __restrict__ inv_in,
            const float* __restrict__ rsq_in, float* __restrict__ out) {
    __shared__ float lds_w[DD * WSTRIDE];   // 20 KB: 64 rows x 80-dword stride

    int wid = threadIdx.x >> 5;
    int lane = threadIdx.x & 31;

#if defined(__gfx1250__) && __has_builtin(__builtin_amdgcn_tensor_load_to_lds)
    if (wid == 0) {
        // TDM descriptor: 2-D tensor, 64x64 tile of 4-byte elements, contiguous in global,
        // LDS padding of 16 dwords after every 64 dwords (row stride 80).
        unsigned long long ga = (unsigned long long)(const void*)W;
        // Generic pointers to LDS carry the byte offset in the low 32 bits.
        unsigned lds_base = (unsigned)(unsigned long long)(const void*)lds_w;
        v4u g0;
        g0.x = 1u;                                     // count=1 (valid user descriptor)
        g0.y = lds_base;                               // lds_addr (bytes)
        g0.z = (unsigned)(ga & 0xFFFFFFFFu);           // global_addr[31:0]
        g0.w = (unsigned)((ga >> 32) & 0x01FFFFFFu)    // global_addr[56:32]
             | 0x80000000u;                            // type = 2 ("image")
        v8i g1;
        g1[0] = 0x1F520000;          // pad_amount=15(16dw), pad_interval=5(64dw),
                                     // pad_enable=1, data_size=2(4B), wg_mask=0
        g1[1] = (int)(64u << 16);    // tensor_dim0 = 64 (low 16 bits)
        g1[2] = (int)(64u << 16);    // tensor_dim1 = 64
        g1[3] = (int)(64u << 16);    // tile_dim0 = 64
        g1[4] = 64;                  // tile_dim1 = 64; tile_dim2 = 0
        g1[5] = 64;                  // tensor_dim0_stride = 64 (low 32 bits)
        g1[6] = (int)(64u << 16);    // dim0_stride hi = 0; tensor_dim1_stride lo = 64
        g1[7] = 0;                   // tensor_dim1_stride hi = 0
        v4i z4 = {0, 0, 0, 0};       // groups 2/3 unused (<=2D tensor)
        v8i z8 = {0, 0, 0, 0, 0, 0, 0, 0};
        __builtin_amdgcn_tensor_load_to_lds(g0, g1, z4, z4, z8, 0);
        __builtin_amdgcn_s_wait_tensorcnt(0);
    }
#else
    for (int i = threadIdx.x; i < DD * DD; i += 256)
        lds_w[(i >> 6) * WSTRIDE + (i & 63)] = W[i];
#endif
    __syncthreads();
    // The TDM builtin has no pointer-based dependence on lds_w; without this clobber the
    // compiler sees "no stores to lds_w" and folds the B-fragment loads to undef.
    asm volatile("" : : : "memory");

    int wave = blockIdx.x * (blockDim.x >> 5) + wid;
    int row_base = wave * 16;
    if (row_base >= NN) return;           // wave-uniform exit: EXEC stays all-1s for WMMA

    int m = lane & 15;                    // row (A) / col (B,C,D) within tile
    int hi = lane >> 4;                   // 0: lanes 0-15, 1: lanes 16-31

    v8f c[4] = {};                        // 4 N-tiles of 16x16 f32 accumulators

    for (int kt = 0; kt < 16; ++kt) {
        int k0 = kt * 4;
        // A fragment 16x4: lane m holds (row=m, K = k0 + 2*hi + {0,1}) -> contiguous float2
        v2f a = *(const v2f*)(x + (size_t)(row_base + m) * DD + k0 + 2 * hi);
#pragma unroll
        for (int nt = 0; nt < 4; ++nt) {
            int n0 = nt * 16;
            // B fragment 4x16 from LDS: v0 row K=k0+hi, v1 row K=k0+2+hi, column n0+m.
            // Row stride 80 dwords -> lanes 0-15 and 16-31 hit disjoint bank sets.
            v2f b;
            b.x = lds_w[(k0 + hi) * WSTRIDE + n0 + m];
            b.y = lds_w[(k0 + 2 + hi) * WSTRIDE + n0 + m];
            c[nt] = __builtin_amdgcn_wmma_f32_16x16x4_f32(
                false, a, false, b, (short)0, c[nt], false, false);
        }
    }

    // Epilogue: C/D layout -> vgpr r: row = r + 8*hi, col = m (within each 16-col tile)
#pragma unroll
    for (int r = 0; r < 8; ++r) {
        int row = row_base + 8 * hi + r;
        float ii = inv_in[row];
        float rs = rsq_in[row];
        const float* mrow = msum + (size_t)row * DD;
        float* orow = out + (size_t)row * DD;
#pragma unroll
        for (int nt = 0; nt < 4; ++nt) {
            int col = nt * 16 + m;
            float v = (c[nt][r] + mrow[col] * ii) * rs;
            orow[col] = fmaxf(v, 0.0f);
        }
    }
}

extern "C" void kernel_launch(void* const* d_in, const int* in_sizes, int n_in,
                              void* d_out, int out_size, void* d_ws, size_t ws_size,
                              hipStream_t stream) {
    const float* x   = (const float*)d_in[0];
    const float* rel = (const float*)d_in[1];
    const float* w1  = (const float*)d_in[2];
    const float* w2  = (const float*)d_in[3];
    const int*   src = (const int*)d_in[4];
    const int*   dst = (const int*)d_in[5];
    const int*   et  = (const int*)d_in[6];
    float* out = (float*)d_out;

    float* ws       = (float*)d_ws;
    float* deg_out  = ws;                         // [NN]
    float* deg_in   = ws + (size_t)NN;            // [NN]
    float* norm_out = ws + (size_t)2 * NN;        // [NN]
    float* inv_in   = ws + (size_t)3 * NN;        // [NN]
    float* rsq_in   = ws + (size_t)4 * NN;        // [NN]
    float* msum     = ws + (size_t)5 * NN;        // [NN*DD]
    float* h1       = ws + (size_t)5 * NN + (size_t)NN * DD; // [NN*DD]

    const int T = 256;
    const int ND = NN * DD;
    const int edge_blocks = (NE + 7) / 8;                 // 8 waves/block, 1 wave/edge
    const int node_blocks = ((NN + 15) / 16 + 7) / 8;     // 8 waves/block, 16 rows/wave

    // Degrees + norm factors (shared by both layers)
    zero_kernel<<<(2 * NN + T - 1) / T, T, 0, stream>>>(deg_out, 2 * NN);
    degree_kernel<<<(NE + T - 1) / T, T, 0, stream>>>(src, dst, deg_out, deg_in);
    norm_kernel<<<(NN + T - 1) / T, T, 0, stream>>>(deg_out, deg_in, norm_out, inv_in, rsq_in);

    // Layer 1: x -> h1
    zero_kernel<<<(ND + T - 1) / T, T, 0, stream>>>(msum, ND);
    edge_kernel<<<edge_blocks, T, 0, stream>>>(x, rel, src, dst, et, norm_out, msum);
    node_kernel<<<node_blocks, T, 0, stream>>>(x, w1, msum, inv_in, rsq_in, h1);

    // Layer 2: h1 -> out
    zero_kernel<<<(ND + T - 1) / T, T, 0, stream>>>(msum, ND);
    edge_kernel<<<edge_blocks, T, 0, stream>>>(h1, rel, src, dst, et, norm_out, msum);
    node_kernel<<<node_blocks, T, 0, stream>>>(h1, w2, msum, inv_in, rsq_in, out);
}